// GATPolicy_57037165691499
// MI455X (gfx1250) — compile-verified
//
#include <hip/hip_runtime.h>
#include <math.h>

// ---------------------------------------------------------------------------
// GAT (PyG GATConv-style), 8 layers, on MI455X (gfx1250, wave32, WMMA).
// GEMM uses V_WMMA_F32_16X16X4_F32 (fp32 in/out, matches reference precision).
// LDS layouts swizzled/padded so every WMMA fragment is a single bank-conflict-
// free ds_load_b64 into an even-aligned VGPR pair (no v_mov reassembly).
// ---------------------------------------------------------------------------

typedef __attribute__((ext_vector_type(2))) float v2f;
typedef __attribute__((ext_vector_type(8))) float v8f;

#define NB   16
#define NN   10000
#define NF   64
#define NH   8
#define ND   8
#define NE   160000
#define NEP  (NE + NN)          // edges + self loops = 170000
#define NLAY 8
#define SLOPE 0.2f

#define TILES (NN / 16)         // 625 row tiles per graph (N divisible by 16)
#define WPB   8                 // waves per GEMM block (256 threads)
#define BPG   ((TILES + WPB - 1) / WPB)   // 79 blocks per graph

#define SX_STRIDE 68            // padded x-tile row stride (floats): bank = 4*rc+k
#define SW_STRIDE 160           // W pair-row stride (floats): +160 == +32 banks

// order-preserving float -> uint mapping for atomic max (key 0 == -inf)
__device__ __forceinline__ unsigned fkey(float f) {
    unsigned u = __float_as_uint(f);
    return (u & 0x80000000u) ? ~u : (u | 0x80000000u);
}
__device__ __forceinline__ float unfkey(unsigned k) {
    return (k & 0x80000000u) ? __uint_as_float(k ^ 0x80000000u)
                             : __uint_as_float(~k);
}

// ---------------------------------------------------------------------------
// Kernel 1: h[b] = x[b] @ W  via v_wmma_f32_16x16x4_f32.
// Each wave computes a 16x64 output tile: 4 accumulators x 16 K-steps.
// ---------------------------------------------------------------------------
__global__ __launch_bounds__(256) void gat_gemm_kernel(
    const float* __restrict__ x, const float* __restrict__ W,
    float* __restrict__ h)
{
    // W swizzled by k-pairs: element W[k][c] at sW[(k>>1)*SW_STRIDE + c*2 + (k&1)]
    __shared__ __align__(16) float sW[32 * SW_STRIDE];            // 20 KB
    __shared__ __align__(16) float sX[WPB][16 * SX_STRIDE];       // ~34 KB

    const int tid  = threadIdx.x;
    const int wave = tid >> 5;
    const int lane = tid & 31;
    const int b    = blockIdx.x / BPG;
    const int tblk = blockIdx.x % BPG;
    const int tile = tblk * WPB + wave;

    // cooperative swizzled load of W (64x64 fp32) into LDS
    {
        const float4* Wv = (const float4*)W;
#pragma unroll
        for (int i = 0; i < 4; ++i) {
            const int idx4 = tid + i * 256;
            const float4 w4 = Wv[idx4];
            const int j0 = idx4 * 4;            // linear float index
            const int k  = j0 >> 6;             // W row
            const int c  = j0 & 63;             // W col (multiple of 4)
            float* dstp = &sW[(k >> 1) * SW_STRIDE + c * 2 + (k & 1)];
            dstp[0] = w4.x; dstp[2] = w4.y; dstp[4] = w4.z; dstp[6] = w4.w;
        }
    }
    // per-wave load of its contiguous 16x64 x tile into padded LDS rows
    if (tile < TILES) {
        const float4* xv =
            (const float4*)(x + ((size_t)b * NN + (size_t)tile * 16) * NF);
#pragma unroll
        for (int i = 0; i < 8; ++i) {
            const int idx4 = lane + i * 32;
            const float4 v = xv[idx4];
            const int j0 = idx4 * 4;
            const int r  = j0 >> 6;
            const int c  = j0 & 63;
            *(float4*)&sX[wave][r * SX_STRIDE + c] = v;
        }
    }
    __syncthreads();
    if (tile >= TILES) return;

    const int rc = lane & 15;       // M-row (A frag) / N-col (B,C,D frags)
    const int kh = lane >> 4;       // lane half selects K sub-pair

    v8f acc0 = {}, acc1 = {}, acc2 = {}, acc3 = {};
    const float* sx = sX[wave];

#pragma unroll
    for (int t = 0; t < 16; ++t) {
        const int k0 = 4 * t + kh * 2;
        // A[16x4]: lane=M, vgpr=K -> contiguous 8B LDS load
        const v2f a = *(const v2f*)&sx[rc * SX_STRIDE + k0];
        // B[4x16]: lane=N, vgpr=K -> contiguous 8B loads from swizzled W
        const float* bp = &sW[(2 * t + kh) * SW_STRIDE + rc * 2];
        const v2f b0 = *(const v2f*)&bp[0];
        const v2f b1 = *(const v2f*)&bp[32];
        const v2f b2 = *(const v2f*)&bp[64];
        const v2f b3 = *(const v2f*)&bp[96];
        acc0 = __builtin_amdgcn_wmma_f32_16x16x4_f32(false, a, false, b0,
                                                     (short)0, acc0, false, false);
        acc1 = __builtin_amdgcn_wmma_f32_16x16x4_f32(false, a, false, b1,
                                                     (short)0, acc1, false, false);
        acc2 = __builtin_amdgcn_wmma_f32_16x16x4_f32(false, a, false, b2,
                                                     (short)0, acc2, false, false);
        acc3 = __builtin_amdgcn_wmma_f32_16x16x4_f32(false, a, false, b3,
                                                     (short)0, acc3, false, false);
    }

    // C/D layout: vgpr r, lane l -> row = r + 8*(l>>4), col = l&15
    float* hb = h + ((size_t)b * NN + (size_t)tile * 16 + (size_t)(lane >> 4) * 8) * NF
                  + (lane & 15);
#pragma unroll
    for (int r = 0; r < 8; ++r) {
        hb[r * NF +  0] = acc0[r];
        hb[r * NF + 16] = acc1[r];
        hb[r * NF + 32] = acc2[r];
        hb[r * NF + 48] = acc3[r];
    }
}

// ---------------------------------------------------------------------------
// Kernel 2: per (b,n,head): attention logits al_s/al_d, init max/denom/out.
// ---------------------------------------------------------------------------
__global__ void gat_al_init_kernel(
    const float* __restrict__ h,
    const float* __restrict__ a_s, const float* __restrict__ a_d,
    float* __restrict__ al_s, float* __restrict__ al_d,
    unsigned* __restrict__ mkey, float* __restrict__ denom,
    float* __restrict__ outacc)
{
    const int tid = blockIdx.x * blockDim.x + threadIdx.x;
    if (tid >= NB * NN * NH) return;
    const int head = tid & (NH - 1);
    const int bn   = tid >> 3;          // b*NN + n
    const float* hp = h + (size_t)bn * NF + head * ND;
    float s = 0.f, d = 0.f;
#pragma unroll
    for (int k = 0; k < ND; ++k) {
        const float v = hp[k];
        s += v * a_s[head * ND + k];
        d += v * a_d[head * ND + k];
    }
    al_s[tid]  = s;
    al_d[tid]  = d;
    mkey[tid]  = 0u;     // == -inf under fkey ordering
    denom[tid] = 0.f;
    float* op = outacc + (size_t)bn * NF + head * ND;
#pragma unroll
    for (int k = 0; k < ND; ++k) op[k] = 0.f;
}

// ---------------------------------------------------------------------------
// Kernel 3: per (b,edge): segment max of leaky-relu logits (atomic u32 max).
// ---------------------------------------------------------------------------
__global__ void gat_edge_max_kernel(
    const int* __restrict__ src_e, const int* __restrict__ dst_e,
    const float* __restrict__ al_s, const float* __restrict__ al_d,
    unsigned* __restrict__ mkey)
{
    const int tid = blockIdx.x * blockDim.x + threadIdx.x;
    if (tid >= NB * NEP) return;
    const int e = tid % NEP;
    const int b = tid / NEP;
    const int s = (e < NE) ? src_e[e] : (e - NE);
    const int d = (e < NE) ? dst_e[e] : (e - NE);
    const float* as = al_s + ((size_t)b * NN + s) * NH;
    const float* ad = al_d + ((size_t)b * NN + d) * NH;
    unsigned*    mk = mkey + ((size_t)b * NN + d) * NH;
#pragma unroll
    for (int hh = 0; hh < NH; ++hh) {
        float ev = as[hh] + ad[hh];
        ev = (ev > 0.f) ? ev : ev * SLOPE;
        atomicMax(&mk[hh], fkey(ev));
    }
}

// ---------------------------------------------------------------------------
// Kernel 4: per (b,edge): denom += exp(e - m[dst]).
// ---------------------------------------------------------------------------
__global__ void gat_edge_sum_kernel(
    const int* __restrict__ src_e, const int* __restrict__ dst_e,
    const float* __restrict__ al_s, const float* __restrict__ al_d,
    const unsigned* __restrict__ mkey, float* __restrict__ denom)
{
    const int tid = blockIdx.x * blockDim.x + threadIdx.x;
    if (tid >= NB * NEP) return;
    const int e = tid % NEP;
    const int b = tid / NEP;
    const int s = (e < NE) ? src_e[e] : (e - NE);
    const int d = (e < NE) ? dst_e[e] : (e - NE);
    const float*    as = al_s  + ((size_t)b * NN + s) * NH;
    const float*    ad = al_d  + ((size_t)b * NN + d) * NH;
    const unsigned* mk = mkey  + ((size_t)b * NN + d) * NH;
    float*          dn = denom + ((size_t)b * NN + d) * NH;
#pragma unroll
    for (int hh = 0; hh < NH; ++hh) {
        float ev = as[hh] + ad[hh];
        ev = (ev > 0.f) ? ev : ev * SLOPE;
        atomicAdd(&dn[hh], __expf(ev - unfkey(mk[hh])));
    }
}

// ---------------------------------------------------------------------------
// Kernel 5: per (b,edge,head): out[dst] += alpha * h[src] (8 fp32 atomics).
// ---------------------------------------------------------------------------
__global__ void gat_edge_scatter_kernel(
    const int* __restrict__ src_e, const int* __restrict__ dst_e,
    const float* __restrict__ al_s, const float* __restrict__ al_d,
    const unsigned* __restrict__ mkey, const float* __restrict__ denom,
    const float* __restrict__ h, float* __restrict__ outacc)
{
    const int tid = blockIdx.x * blockDim.x + threadIdx.x;
    if (tid >= NB * NEP * NH) return;
    const int head = tid & (NH - 1);
    const int et   = tid >> 3;
    const int e    = et % NEP;
    const int b    = et / NEP;
    const int s = (e < NE) ? src_e[e] : (e - NE);
    const int d = (e < NE) ? dst_e[e] : (e - NE);

    float ev = al_s[((size_t)b * NN + s) * NH + head] +
               al_d[((size_t)b * NN + d) * NH + head];
    ev = (ev > 0.f) ? ev : ev * SLOPE;
    const float m   = unfkey(mkey[((size_t)b * NN + d) * NH + head]);
    const float den = denom[((size_t)b * NN + d) * NH + head];
    const float alpha = __expf(ev - m) / den;

    const float* hp = h      + ((size_t)b * NN + s) * NF + head * ND;
    float*       op = outacc + ((size_t)b * NN + d) * NF + head * ND;
#pragma unroll
    for (int k = 0; k < ND; ++k)
        atomicAdd(&op[k], hp[k] * alpha);
}

// ---------------------------------------------------------------------------
// Kernel 6: x_next = out + bias  (last layer writes d_out).
// ---------------------------------------------------------------------------
__global__ void gat_bias_kernel(
    const float* __restrict__ outacc, const float* __restrict__ bias,
    float* __restrict__ xnext)
{
    const int tid = blockIdx.x * blockDim.x + threadIdx.x;
    if (tid >= NB * NN * NF) return;
    xnext[tid] = outacc[tid] + bias[tid & (NF - 1)];
}

// ---------------------------------------------------------------------------
extern "C" void kernel_launch(void* const* d_in, const int* in_sizes, int n_in,
                              void* d_out, int out_size, void* d_ws, size_t ws_size,
                              hipStream_t stream)
{
    (void)in_sizes; (void)n_in; (void)out_size; (void)ws_size;

    const float* x     = (const float*)d_in[0];   // [B,N,F]
    const float* Ws    = (const float*)d_in[1];   // [L,F,F]
    const float* att_s = (const float*)d_in[2];   // [L,H,D]
    const float* att_d = (const float*)d_in[3];   // [L,H,D]
    const float* bias  = (const float*)d_in[4];   // [L,F]
    const int*   ei    = (const int*)d_in[5];     // [2,E]
    const int* src_e = ei;
    const int* dst_e = ei + NE;

    float* ws = (float*)d_ws;
    const size_t big   = (size_t)NB * NN * NF;    // 10.24M floats
    const size_t small = (size_t)NB * NN * NH;    // 1.28M floats
    float*    buf_x   = ws;
    float*    buf_h   = ws + big;
    float*    buf_out = ws + 2 * big;
    float*    al_s    = ws + 3 * big;
    float*    al_d    = al_s + small;
    unsigned* mkey    = (unsigned*)(al_d + small);
    float*    denom   = (float*)(mkey + small);

    const int n_bnh = NB * NN * NH;
    const int n_be  = NB * NEP;
    const int n_beh = NB * NEP * NH;
    const int n_bnf = NB * NN * NF;

    for (int l = 0; l < NLAY; ++l) {
        const float* xin  = (l == 0) ? x : buf_x;
        float*       xout = (l == NLAY - 1) ? (float*)d_out : buf_x;
        const float* W  = Ws    + (size_t)l * NF * NF;
        const float* as = att_s + (size_t)l * NH * ND;
        const float* ad = att_d + (size_t)l * NH * ND;
        const float* bi = bias  + (size_t)l * NF;

        gat_gemm_kernel<<<NB * BPG, 256, 0, stream>>>(xin, W, buf_h);
        gat_al_init_kernel<<<(n_bnh + 255) / 256, 256, 0, stream>>>(
            buf_h, as, ad, al_s, al_d, mkey, denom, buf_out);
        gat_edge_max_kernel<<<(n_be + 255) / 256, 256, 0, stream>>>(
            src_e, dst_e, al_s, al_d, mkey);
        gat_edge_sum_kernel<<<(n_be + 255) / 256, 256, 0, stream>>>(
            src_e, dst_e, al_s, al_d, mkey, denom);
        gat_edge_scatter_kernel<<<(n_beh + 255) / 256, 256, 0, stream>>>(
            src_e, dst_e, al_s, al_d, mkey, denom, buf_h, buf_out);
        gat_bias_kernel<<<(n_bnf + 255) / 256, 256, 0, stream>>>(buf_out, bi, xout);
    }
}